// ImprovedContrastiveLoss_89988154786575
// MI455X (gfx1250) — compile-verified
//
#include <hip/hip_runtime.h>
#include <math.h>

typedef __attribute__((ext_vector_type(2))) float v2f;
typedef __attribute__((ext_vector_type(8))) float v8f;

#define MARGIN 0.3f
#define ANCHOR 0.5f
#define CEPS   1e-8f
#define TJ     1024

// ---------------------------------------------------------------- zero accumulators
__global__ void icl_zero_acc(double* total, unsigned long long* cnt) {
    *total = 0.0;
    *cnt   = 0ull;
}

// ---------------------------------------------------------------- phase 1: s_q + quality
// One wave (32 threads) per 16-row tile. Dot products via V_WMMA_F32_16X16X4_F32
// with q broadcast across all 16 B columns; row sum-of-squares via per-lane FMA
// on the same A operands.
__global__ __launch_bounds__(32) void icl_sq_kernel(
    const float* __restrict__ d_embs,
    const float* __restrict__ q,
    const float* __restrict__ rates,
    float* __restrict__ s_q,
    float* __restrict__ quality,
    int N, int D)
{
    const int lane = threadIdx.x;        // 0..31 (wave32)
    const int tile = blockIdx.x;         // N/16 tiles
    const int row  = lane & 15;          // A-matrix M for this lane
    const int half = lane >> 4;          // 0: K={k,k+1}, 1: K={k+2,k+3}

    // ||q||^2 : strided per-lane partial, then butterfly reduce across the wave
    float qss = 0.f;
    for (int k = lane; k < D; k += 32) { float v = q[k]; qss += v * v; }
#pragma unroll
    for (int m = 16; m >= 1; m >>= 1) qss += __shfl_xor(qss, m, 32);
    const float qnorm = sqrtf(qss);

    const float* rowp = d_embs + (size_t)(tile * 16 + row) * (size_t)D;

    v8f c = {0.f, 0.f, 0.f, 0.f, 0.f, 0.f, 0.f, 0.f};
    float ss = 0.f;

    for (int k = 0; k < D; k += 4) {
        // A 16x4 f32 layout: lanes 0-15 hold K={k,k+1} of row M=lane,
        //                    lanes 16-31 hold K={k+2,k+3} of row M=lane-16
        v2f a = *(const v2f*)(rowp + k + 2 * half);
        // B 4x16 with every column == q  ->  b per lane: {q[k+half], q[k+2+half]}
        v2f b;
        b.x = q[k + half];
        b.y = q[k + 2 + half];

        ss += a.x * a.x + a.y * a.y;   // row-norm partial (rides along in VALU)

        // D = A x B + C   (emits v_wmma_f32_16x16x4_f32)
        c = __builtin_amdgcn_wmma_f32_16x16x4_f32(
                /*neg_a=*/false, a, /*neg_b=*/false, b,
                /*c_mod=*/(short)0, c, /*reuse_a=*/false, /*reuse_b=*/false);
    }

    // combine the two K-halves of each row's sum-of-squares
    ss += __shfl_xor(ss, 16, 32);

    __shared__ float ssh[16];
    if (half == 0) ssh[row] = ss;      // lanes 0-15 hold rows 0-15 now
    __syncthreads();

    // C/D layout: lanes 0-15: c[v] = D[v][lane]      (all columns identical)
    //             lanes 16-31: c[v] = D[v+8][lane-16]
    if (lane == 0 || lane == 16) {
#pragma unroll
        for (int v = 0; v < 8; ++v) {
            int r = v + 8 * half;
            float dot = c[v];
            float den = fmaxf(sqrtf(ssh[r]) * qnorm, CEPS);
            s_q[tile * 16 + r] = dot / den;
        }
    }
    if (half == 0) {
        int i = tile * 16 + row;
        quality[i] = fabsf(rates[i] - ANCHOR);
    }
}

// ---------------------------------------------------------------- phase 2: pairwise loss
// Block = 256 threads = 8 waves. Thread owns one i; j-chunk of TJ staged in LDS.
__global__ __launch_bounds__(256) void icl_pair_kernel(
    const float* __restrict__ s_q,
    const float* __restrict__ quality,
    double* __restrict__ total,
    unsigned long long* __restrict__ cnt,
    int N)
{
    __shared__ float sj[TJ];
    __shared__ float qj[TJ];

    const int tid = threadIdx.x;
    const int i   = blockIdx.x * 256 + tid;
    const int j0  = blockIdx.y * TJ;

    float si = 0.f, qi = 0.f;
    const bool valid = (i < N);
    if (valid) { si = s_q[i]; qi = quality[i]; }

    for (int t = tid; t < TJ; t += 256) {
        int j = j0 + t;
        // OOB j gets quality = +inf-ish so qi > qj is never true
        sj[t] = (j < N) ? s_q[j]     : 0.f;
        qj[t] = (j < N) ? quality[j] : 3.0e38f;
    }
    __syncthreads();

    float acc = 0.f;
    int   pc  = 0;
    if (valid) {
        for (int t = 0; t < TJ; ++t) {
            float pl = MARGIN + sj[t] - si;
            if (qi > qj[t]) {
                pc  += 1;
                acc += fmaxf(pl, 0.f);
            }
        }
    }

    // wave-level butterfly reduce, then LDS across the 8 waves
#pragma unroll
    for (int m = 16; m >= 1; m >>= 1) {
        acc += __shfl_xor(acc, m, 32);
        pc  += __shfl_xor(pc,  m, 32);
    }
    __shared__ float wacc[8];
    __shared__ int   wcnt[8];
    const int wave = tid >> 5;
    if ((tid & 31) == 0) { wacc[wave] = acc; wcnt[wave] = pc; }
    __syncthreads();

    if (tid == 0) {
        double bsum = 0.0;
        unsigned long long bcnt = 0ull;
#pragma unroll
        for (int w = 0; w < 8; ++w) {
            bsum += (double)wacc[w];
            bcnt += (unsigned long long)(unsigned int)wcnt[w];
        }
        atomicAdd(total, bsum);                           // global_atomic_add_f64
        atomicAdd(cnt, bcnt);                             // global_atomic_add_u64
    }
}

// ---------------------------------------------------------------- finalize
__global__ void icl_finalize(const double* __restrict__ total,
                             const unsigned long long* __restrict__ cnt,
                             float* __restrict__ out)
{
    unsigned long long n = *cnt;
    if (n < 1ull) n = 1ull;
    out[0] = (float)(*total / (double)n);
}

// ---------------------------------------------------------------- launcher
extern "C" void kernel_launch(void* const* d_in, const int* in_sizes, int n_in,
                              void* d_out, int out_size, void* d_ws, size_t ws_size,
                              hipStream_t stream)
{
    const float* q_emb  = (const float*)d_in[0];   // [D]
    const float* d_embs = (const float*)d_in[1];   // [N, D]
    // d_in[2] = c_emb : dead code in the reference (s_c is never used) -> skipped
    const float* rates  = (const float*)d_in[3];   // [N]

    const int D = in_sizes[0];
    const int N = in_sizes[3];

    // workspace layout: s_q[N] f32 | quality[N] f32 | (8B aligned) total f64 | cnt u64
    float* s_q     = (float*)d_ws;
    float* quality = s_q + N;
    size_t off     = ((size_t)2 * (size_t)N * sizeof(float) + 7ull) & ~7ull;
    double* total  = (double*)((char*)d_ws + off);
    unsigned long long* cnt = (unsigned long long*)(total + 1);

    float* out = (float*)d_out;

    icl_zero_acc<<<1, 1, 0, stream>>>(total, cnt);

    icl_sq_kernel<<<N / 16, 32, 0, stream>>>(d_embs, q_emb, rates, s_q, quality, N, D);

    dim3 grid((N + 255) / 256, (N + TJ - 1) / TJ);
    icl_pair_kernel<<<grid, 256, 0, stream>>>(s_q, quality, total, cnt, N);

    icl_finalize<<<1, 1, 0, stream>>>(total, cnt, out);
}